// CCPL_14216341750304
// MI455X (gfx1250) — compile-verified
//
#include <hip/hip_runtime.h>
#include <math.h>

typedef float v2f __attribute__((ext_vector_type(2)));
typedef float v8f __attribute__((ext_vector_type(8)));

#define BB   8
#define CCH  64
#define HH   256
#define WWI  256
#define HWSZ (HH * WWI)
#define SSN  2048            // S = 8 * 256
#define TAUF 0.07f
#define EPSN 1e-7f

__global__ void CCPL_zero_kernel(float* out) { out[0] = 0.0f; }

// ---------------------------------------------------------------------------
// Kernel 1: gather-diff + MLP(64->64 relu ->16) + channel L2 norm.
// One wave (32 lanes) per 16-row tile. blockIdx.y selects fq vs fk.
// A-layout (16x4 f32): lane&15 = M (row), lane>>4 selects K pair {0,1}/{2,3}.
// B-layout (4x16 f32): lane&15 = N (col), lane>>4 selects K rows  {0,1}/{2,3}.
// D-layout (16x16 f32): vgpr r, lanes 0-15 -> M=r, lanes 16-31 -> M=r+8; N=lane&15.
// ---------------------------------------------------------------------------
__global__ __launch_bounds__(32) void CCPL_mlp_kernel(
    const float* __restrict__ fq, const float* __restrict__ fk,
    const float* __restrict__ W0, const float* __restrict__ b0,
    const float* __restrict__ W1, const float* __restrict__ b1,
    const long long* __restrict__ c_ids, const long long* __restrict__ n_ids,
    float* __restrict__ outq, float* __restrict__ outk)
{
    const int l    = threadIdx.x;
    const int lm   = l & 15;
    const int half = l >> 4;

    const int tile    = blockIdx.x;        // 0..(B*S/16 - 1)
    const int rowBase = tile * 16;         // flat row = b*S + s
    const int b       = rowBase / SSN;
    const int sBase   = rowBase - b * SSN;

    const float* feat = (blockIdx.y == 0) ? fq : fk;
    float*       out  = (blockIdx.y == 0) ? outq : outk;

    // ---- gather + diff directly into A-matrix register layout ----
    const int       s    = sBase + lm;
    const long long cpix = c_ids[s];
    const long long npix = n_ids[s];
    const float*    fb   = feat + (size_t)b * CCH * HWSZ;

    v2f a16[16];
#pragma unroll
    for (int kk = 0; kk < 16; ++kk) {
        const int ch = 4 * kk + 2 * half;
        const float d0 = fb[(size_t)ch * HWSZ + cpix] - fb[(size_t)ch * HWSZ + npix];
        const float d1 = fb[(size_t)(ch + 1) * HWSZ + cpix] - fb[(size_t)(ch + 1) * HWSZ + npix];
        a16[kk][0] = d0;
        a16[kk][1] = d1;
    }

    // ---- layer 0: H = relu(X @ W0 + b0), 4 N-tiles x 16 K-steps of WMMA ----
    v8f hacc[4];
#pragma unroll
    for (int nt = 0; nt < 4; ++nt) hacc[nt] = (v8f){0,0,0,0,0,0,0,0};

#pragma unroll
    for (int kk = 0; kk < 16; ++kk) {
        const int krow = 4 * kk + 2 * half;
#pragma unroll
        for (int nt = 0; nt < 4; ++nt) {
            v2f bv;
            bv[0] = W0[(size_t)krow       * 64 + nt * 16 + lm];
            bv[1] = W0[(size_t)(krow + 1) * 64 + nt * 16 + lm];
            hacc[nt] = __builtin_amdgcn_wmma_f32_16x16x4_f32(
                false, a16[kk], false, bv, (short)0, hacc[nt], false, false);
        }
    }

    // ---- bias + relu, round-trip through LDS to convert D-layout -> A-layout ----
    __shared__ float Hs[16 * 64];
#pragma unroll
    for (int nt = 0; nt < 4; ++nt) {
        const float bb = b0[nt * 16 + lm];
#pragma unroll
        for (int r = 0; r < 8; ++r) {
            float h = hacc[nt][r] + bb;
            h = h > 0.0f ? h : 0.0f;
            Hs[(r + 8 * half) * 64 + nt * 16 + lm] = h;
        }
    }
    __syncthreads();

    // ---- layer 1: Y = H @ W1 + b1 (single 16-wide N tile, 16 K-steps) ----
    v8f yacc = (v8f){0,0,0,0,0,0,0,0};
#pragma unroll
    for (int kk = 0; kk < 16; ++kk) {
        const int ch = 4 * kk + 2 * half;
        v2f av, bv;
        av[0] = Hs[lm * 64 + ch];
        av[1] = Hs[lm * 64 + ch + 1];
        bv[0] = W1[ch * 16 + lm];
        bv[1] = W1[(ch + 1) * 16 + lm];
        yacc = __builtin_amdgcn_wmma_f32_16x16x4_f32(
            false, av, false, bv, (short)0, yacc, false, false);
    }

    // ---- bias + L2 normalize over the 16 output channels (across 16 lanes) ----
    const float bb1 = b1[lm];
#pragma unroll
    for (int r = 0; r < 8; ++r) {
        const float y = yacc[r] + bb1;
        float sq = y * y;
        sq += __shfl_xor(sq, 1, 32);
        sq += __shfl_xor(sq, 2, 32);
        sq += __shfl_xor(sq, 4, 32);
        sq += __shfl_xor(sq, 8, 32);
        const float yn = y / (sqrtf(sq) + EPSN);
        out[(size_t)(rowBase + r + 8 * half) * 16 + lm] = yn;
    }
}

// ---------------------------------------------------------------------------
// Kernel 2: fused PatchNCE. Per wave: 16 query rows, stream 128 fk tiles,
// 4x WMMA f32 16x16x4 per tile (K=16), exp/accumulate in-register (no max
// subtraction needed: |logit| <= 1/tau ~ 14.3, exp sums stay < 4e9 in f32).
// ---------------------------------------------------------------------------
__global__ __launch_bounds__(32) void CCPL_nce_kernel(
    const float* __restrict__ fqn, const float* __restrict__ fkn,
    float* __restrict__ loss)
{
    const int l    = threadIdx.x;
    const int lm   = l & 15;
    const int half = l >> 4;

    const int idx = blockIdx.x;          // 0..(B * S/16 - 1)
    const int b   = idx >> 7;            // S/16 = 128
    const int s0  = (idx & 127) * 16;
    const float invTau = 1.0f / TAUF;

    const float* fqb = fqn + (size_t)b * SSN * 16;
    const float* fkb = fkn + (size_t)b * SSN * 16;

    // fq A-tile, resident for the whole t-loop
    v2f aA[4];
#pragma unroll
    for (int kk = 0; kk < 4; ++kk) {
        const float* p = fqb + (size_t)(s0 + lm) * 16 + 4 * kk + 2 * half;
        aA[kk][0] = p[0];
        aA[kk][1] = p[1];
    }

    float acc[8] = {0, 0, 0, 0, 0, 0, 0, 0};

    for (int tt = 0; tt < 128; ++tt) {
        const int t0 = tt * 16;
        v2f bB[4];
#pragma unroll
        for (int kk = 0; kk < 4; ++kk) {
            const float* p = fkb + (size_t)(t0 + lm) * 16 + 4 * kk + 2 * half;
            bB[kk][0] = p[0];
            bB[kk][1] = p[1];
        }
        v8f d = (v8f){0,0,0,0,0,0,0,0};
#pragma unroll
        for (int kk = 0; kk < 4; ++kk) {
            d = __builtin_amdgcn_wmma_f32_16x16x4_f32(
                false, aA[kk], false, bB[kk], (short)0, d, false, false);
        }
#pragma unroll
        for (int r = 0; r < 8; ++r) {
            const int sg = s0 + r + 8 * half;
            const int tg = t0 + lm;
            const float e = __expf(d[r] * invTau);
            acc[r] += (sg == tg) ? 0.0f : e;   // diagonal -> -inf -> exp = 0
        }
    }

    // per-row negative-sum reduction across the 16 lanes sharing a row
    __shared__ float negs[16];
#pragma unroll
    for (int r = 0; r < 8; ++r) {
        float v = acc[r];
        v += __shfl_xor(v, 1, 32);
        v += __shfl_xor(v, 2, 32);
        v += __shfl_xor(v, 4, 32);
        v += __shfl_xor(v, 8, 32);
        if (lm == 0) negs[r + 8 * half] = v;
    }
    __syncthreads();

    // l_pos for row m = lane&15 (both halves compute; upper half zeroed below)
    const float* q = fqb + (size_t)(s0 + lm) * 16;
    const float* k = fkb + (size_t)(s0 + lm) * 16;
    float dot = 0.0f;
#pragma unroll
    for (int c = 0; c < 16; ++c) dot += q[c] * k[c];

    const float posT  = dot * invTau;
    const float total = __expf(posT) + negs[lm];
    float contrib = logf(total) - posT;
    contrib = (l < 16) ? contrib : 0.0f;

    contrib += __shfl_xor(contrib, 1, 32);
    contrib += __shfl_xor(contrib, 2, 32);
    contrib += __shfl_xor(contrib, 4, 32);
    contrib += __shfl_xor(contrib, 8, 32);
    contrib += __shfl_xor(contrib, 16, 32);
    if (l == 0) atomicAdd(loss, contrib * (1.0f / (float)(BB * SSN)));
}

extern "C" void kernel_launch(void* const* d_in, const int* in_sizes, int n_in,
                              void* d_out, int out_size, void* d_ws, size_t ws_size,
                              hipStream_t stream) {
    (void)in_sizes; (void)n_in; (void)out_size; (void)ws_size;

    const float*     f_q   = (const float*)d_in[0];
    const float*     f_k   = (const float*)d_in[1];
    const float*     W0    = (const float*)d_in[2];
    const float*     b0    = (const float*)d_in[3];
    const float*     W1    = (const float*)d_in[4];
    const float*     b1    = (const float*)d_in[5];
    const long long* c_ids = (const long long*)d_in[6];
    const long long* n_ids = (const long long*)d_in[7];

    float* out = (float*)d_out;
    float* fqn = (float*)d_ws;                       // [B*S, 16]
    float* fkn = fqn + (size_t)BB * SSN * 16;        // [B*S, 16]

    CCPL_zero_kernel<<<1, 1, 0, stream>>>(out);

    dim3 gridM(BB * SSN / 16, 2);                    // 1024 tiles x {fq, fk}
    CCPL_mlp_kernel<<<gridM, 32, 0, stream>>>(f_q, f_k, W0, b0, W1, b1,
                                              c_ids, n_ids, fqn, fkn);

    CCPL_nce_kernel<<<BB * SSN / 16, 32, 0, stream>>>(fqn, fkn, out);
}